// DeformationNetworkGraphConvolutional_44555990729275
// MI455X (gfx1250) — compile-verified
//
#include <hip/hip_runtime.h>
#include <hip/hip_bf16.h>
#include <math.h>

// ---------------------------------------------------------------------------
// CDNA5 (gfx1250) implementation of a 7-layer graph-convolutional network.
//   x <- relu( x@W0^T + b0 + scatter_add_over_edges( x@W1^T + b1 ) )
//   out = tanh( x@Woff^T + boff )
// GEMMs use V_WMMA_F32_16X16X4_F32 (exact fp32 matrix math).
// ---------------------------------------------------------------------------

typedef float v2f __attribute__((ext_vector_type(2)));
typedef float v8f __attribute__((ext_vector_type(8)));

#define HID 128

// ---------------------------------------------------------------------------
// Layer 0: din = 3, trivial per-element kernel. t = v*128 + c.
// Writes acc = x@w0^T + b0 and y1 = x@w1^T + b1.
// ---------------------------------------------------------------------------
__global__ __launch_bounds__(256) void gcn_layer0(
    const float* __restrict__ verts,
    const float* __restrict__ w0, const float* __restrict__ b0,
    const float* __restrict__ w1, const float* __restrict__ b1,
    float* __restrict__ acc, float* __restrict__ y1, int V)
{
    int t = blockIdx.x * 256 + threadIdx.x;
    int v = t >> 7;
    int c = t & (HID - 1);
    if (v >= V) return;
    float x0 = verts[v * 3 + 0];
    float x1 = verts[v * 3 + 1];
    float x2 = verts[v * 3 + 2];
    acc[t] = fmaf(x0, w0[c * 3 + 0], fmaf(x1, w0[c * 3 + 1], fmaf(x2, w0[c * 3 + 2], b0[c])));
    y1[t]  = fmaf(x0, w1[c * 3 + 0], fmaf(x1, w1[c * 3 + 1], fmaf(x2, w1[c * 3 + 2], b1[c])));
}

// ---------------------------------------------------------------------------
// Hidden layers 1..6: dual 16x128x128 GEMM per 16-row tile via WMMA f32.
// Workgroup = 256 threads = 8 waves; wave w owns output columns [16w,16w+16).
// x tile (16x128 f32, ReLU fused on load) staged once in LDS, shared by both
// GEMMs of all 8 waves. Weights stream from global (L2-resident, 64 KB each).
//
// Fragment layouts per CDNA5 ISA 7.12.2 (wave32):
//   A (16x4):  lane<16 -> M=lane,   {K, K+1};  lane>=16 -> M=lane-16, {K+2, K+3}
//   B (4x16):  lane<16 -> N=lane,   rows {K, K+1}; lane>=16 -> N=lane-16, rows {K+2, K+3}
//   C/D 16x16: c[r]: lane<16 -> D[r][lane]; lane>=16 -> D[r+8][lane-16]
// ---------------------------------------------------------------------------
__global__ __launch_bounds__(256) void gcn_gemm2_wmma(
    const float* __restrict__ xin,   // V x 128 pre-activation (ReLU applied here)
    const float* __restrict__ w0, const float* __restrict__ b0,
    const float* __restrict__ w1, const float* __restrict__ b1,
    float* __restrict__ acc,         // out: x@w0^T + b0 (scatter target)
    float* __restrict__ y1)          // out: x@w1^T + b1 (scatter source)
{
    __shared__ float xs[16 * HID];   // 8 KB

    const int tid  = threadIdx.x;
    const int row0 = blockIdx.x * 16;

    // Cooperative tile load, 2048 floats as 512 float4, ReLU fused.
    {
        const float4* src = (const float4*)(xin + (size_t)row0 * HID);
        float4*       dst = (float4*)xs;
#pragma unroll
        for (int it = 0; it < 2; ++it) {
            int idx  = tid + it * 256;
            float4 v = src[idx];
            v.x = fmaxf(v.x, 0.0f); v.y = fmaxf(v.y, 0.0f);
            v.z = fmaxf(v.z, 0.0f); v.w = fmaxf(v.w, 0.0f);
            dst[idx] = v;
        }
    }
    __syncthreads();

    const int wave = tid >> 5;        // 0..7 -> column tile
    const int lane = tid & 31;
    const int half = lane >> 4;       // 0: lanes 0-15, 1: lanes 16-31
    const int l    = lane & 15;
    const int n0   = wave * 16;

    const float* arow = xs + l * HID + half * 2;            // A: xs[l][k + 2*half ...]

#pragma unroll
    for (int g = 0; g < 2; ++g) {
        const float* W  = g ? w1 : w0;
        const float* Bv = g ? b1 : b0;
        const float  bias = Bv[n0 + l];

        v8f c;
#pragma unroll
        for (int r = 0; r < 8; ++r) c[r] = bias;

        const float* brow = W + (size_t)(n0 + l) * HID + half * 2;  // B: w[n][k + 2*half ...]

#pragma unroll 4
        for (int k = 0; k < HID; k += 4) {
            v2f a = *(const v2f*)(arow + k);
            v2f b = *(const v2f*)(brow + k);
            // D = A(16x4, f32) * B(4x16, f32) + C
            c = __builtin_amdgcn_wmma_f32_16x16x4_f32(
                    /*neg_a=*/false, a, /*neg_b=*/false, b,
                    /*c_mod=*/(short)0, c, /*reuse_a=*/false, /*reuse_b=*/false);
        }

        float* out = g ? y1 : acc;
#pragma unroll
        for (int r = 0; r < 8; ++r)
            out[(size_t)(row0 + r + half * 8) * HID + (n0 + l)] = c[r];
    }
}

// ---------------------------------------------------------------------------
// Edge scatter: acc[i] += y1[j]; acc[j] += y1[i].  One wave per edge.
// Each lane owns one float4 (128 floats = 32 float4 per row).
// ---------------------------------------------------------------------------
__global__ __launch_bounds__(256) void gcn_scatter(
    const int* __restrict__ edges,   // E x 2 (int32)
    const float* __restrict__ y1,
    float* __restrict__ acc, int E)
{
    int e    = (blockIdx.x * 256 + threadIdx.x) >> 5;
    int lane = threadIdx.x & 31;
    if (e >= E) return;

    int i = edges[e * 2 + 0];
    int j = edges[e * 2 + 1];

    float4 vi = ((const float4*)(y1 + (size_t)i * HID))[lane];
    float4 vj = ((const float4*)(y1 + (size_t)j * HID))[lane];

    float* ai = acc + (size_t)i * HID + lane * 4;
    float* aj = acc + (size_t)j * HID + lane * 4;

    unsafeAtomicAdd(ai + 0, vj.x);
    unsafeAtomicAdd(ai + 1, vj.y);
    unsafeAtomicAdd(ai + 2, vj.z);
    unsafeAtomicAdd(ai + 3, vj.w);

    unsafeAtomicAdd(aj + 0, vi.x);
    unsafeAtomicAdd(aj + 1, vi.y);
    unsafeAtomicAdd(aj + 2, vi.z);
    unsafeAtomicAdd(aj + 3, vi.w);
}

// ---------------------------------------------------------------------------
// Output head: out = tanh( relu(x) @ w_off^T + b_off ),  w_off is 3x128.
// One wave per vertex; float4 partial dots + wave32 xor reduction.
// ---------------------------------------------------------------------------
__global__ __launch_bounds__(256) void gcn_head(
    const float* __restrict__ x,     // V x 128 pre-activation
    const float* __restrict__ w,     // 3 x 128
    const float* __restrict__ b,     // 3
    float* __restrict__ out, int V)
{
    int v    = (blockIdx.x * 256 + threadIdx.x) >> 5;
    int lane = threadIdx.x & 31;
    if (v >= V) return;

    float4 xv = ((const float4*)(x + (size_t)v * HID))[lane];
    xv.x = fmaxf(xv.x, 0.0f); xv.y = fmaxf(xv.y, 0.0f);
    xv.z = fmaxf(xv.z, 0.0f); xv.w = fmaxf(xv.w, 0.0f);

    float s[3];
#pragma unroll
    for (int m = 0; m < 3; ++m) {
        float4 wv = ((const float4*)(w + m * HID))[lane];
        s[m] = fmaf(xv.x, wv.x, fmaf(xv.y, wv.y, fmaf(xv.z, wv.z, xv.w * wv.w)));
    }
#pragma unroll
    for (int off = 16; off > 0; off >>= 1) {
#pragma unroll
        for (int m = 0; m < 3; ++m) s[m] += __shfl_xor(s[m], off, 32);
    }
    if (lane == 0) {
#pragma unroll
        for (int m = 0; m < 3; ++m) out[(size_t)v * 3 + m] = tanhf(s[m] + b[m]);
    }
}

// ---------------------------------------------------------------------------
// Host orchestration.
// d_in layout (setup_inputs order, params flattened leaf-by-leaf):
//   [0] verts (V*3 f32)   [1] edges (E*2 int)
//   [2+4i..5+4i] = (w0, b0, w1, b1) for layer i = 0..6
//   [30] w_off (3*128)    [31] b_off (3)
// Workspace: accA | accB | y1buf, each V*128 f32.
// ---------------------------------------------------------------------------
extern "C" void kernel_launch(void* const* d_in, const int* in_sizes, int n_in,
                              void* d_out, int out_size, void* d_ws, size_t ws_size,
                              hipStream_t stream) {
    const float* verts = (const float*)d_in[0];
    const int*   edges = (const int*)d_in[1];
    const int V = in_sizes[0] / 3;
    const int E = in_sizes[1] / 2;

    float* bufA = (float*)d_ws;
    float* bufB = bufA + (size_t)V * HID;
    float* y1b  = bufB + (size_t)V * HID;

    const int nLayers = 7;
    float* prev = nullptr;

    for (int l = 0; l < nLayers; ++l) {
        const float* w0 = (const float*)d_in[2 + 4 * l];
        const float* b0 = (const float*)d_in[3 + 4 * l];
        const float* w1 = (const float*)d_in[4 + 4 * l];
        const float* b1 = (const float*)d_in[5 + 4 * l];
        float* acc = (l & 1) ? bufB : bufA;

        if (l == 0) {
            int nthr = V * HID;
            gcn_layer0<<<(nthr + 255) / 256, 256, 0, stream>>>(
                verts, w0, b0, w1, b1, acc, y1b, V);
        } else {
            gcn_gemm2_wmma<<<V / 16, 256, 0, stream>>>(
                prev, w0, b0, w1, b1, acc, y1b);
        }
        gcn_scatter<<<(E * 32 + 255) / 256, 256, 0, stream>>>(edges, y1b, acc, E);
        prev = acc;
    }

    const float* w_off = (const float*)d_in[2 + 4 * nLayers];
    const float* b_off = (const float*)d_in[3 + 4 * nLayers];
    gcn_head<<<(V * 32 + 255) / 256, 256, 0, stream>>>(prev, w_off, b_off,
                                                       (float*)d_out, V);
}